// GCNLayer_15547781612141
// MI455X (gfx1250) — compile-verified
//
#include <hip/hip_runtime.h>
#include <stdint.h>

#define D_FEAT 128

// ---------------------------------------------------------------------------
// Kernel 1: zero the output (harness poisons d_out with 0xAA before timing).
// ---------------------------------------------------------------------------
__global__ void gcn_zero_out(float4* __restrict__ out4, long n4,
                             float* __restrict__ out, long n_total) {
    long i = (long)blockIdx.x * blockDim.x + threadIdx.x;
    long stride = (long)gridDim.x * blockDim.x;
    float4 z = {0.f, 0.f, 0.f, 0.f};
    for (long k = i; k < n4; k += stride) out4[k] = z;
    for (long k = n4 * 4 + i; k < n_total; k += stride) out[k] = 0.f;  // tail
}

// ---------------------------------------------------------------------------
// Kernel 2: edge scatter, software-pipelined.
//
// One wave (32 lanes) per edge; lane l owns features [4l, 4l+3] (wave32 x
// float4 == 128 features exactly). Edge metadata (row/col/val) is staged
// 32-edges-per-wave-chunk into double-buffered LDS with gfx1250 async
// global->LDS copies. Pipeline: issue chunk k+1's 3 async copies, then
// `s_wait_asynccnt 3` — async ops complete in order per wave, so <=3
// outstanding guarantees chunk k is resident while k+1 is in flight.
// Read-back is an explicit ds_load_b32 + s_wait_dscnt (cannot be folded away),
// then per-edge broadcast via v_readlane_b32 into SGPRs.
// ---------------------------------------------------------------------------
__global__ __launch_bounds__(256) void gcn_spmm_scatter(
    const int*   __restrict__ erow,
    const int*   __restrict__ ecol,
    const float* __restrict__ eval,
    const float* __restrict__ embs,
    float*       __restrict__ out,
    int E) {
    // [buf][wave][field(r,c,v)][lane]
    __shared__ int s_meta[2][8][3][32];

    const int lane = threadIdx.x & 31;
    const int wid  = threadIdx.x >> 5;

    const long gw     = (long)blockIdx.x * 8 + wid;
    const long stride = (long)gridDim.x * 8 * 32;

// Stage 3 async b32 copies (row,col,val for edge b+lane) into buffer `buf`.
// Generic->LDS pointer truncation: shared-aperture tag lives in addr[63:32],
// so the low 32 bits are the LDS offset the async engine expects.
#define ISSUE_ASYNC(buf, b)                                                   \
    do {                                                                      \
        const uint64_t gr_ = (uint64_t)(uintptr_t)(erow + (b) + lane);        \
        const uint64_t gc_ = (uint64_t)(uintptr_t)(ecol + (b) + lane);        \
        const uint64_t gv_ = (uint64_t)(uintptr_t)(eval + (b) + lane);        \
        const unsigned lr_ = (unsigned)(uintptr_t)&s_meta[(buf)][wid][0][lane];\
        const unsigned lc_ = (unsigned)(uintptr_t)&s_meta[(buf)][wid][1][lane];\
        const unsigned lv_ = (unsigned)(uintptr_t)&s_meta[(buf)][wid][2][lane];\
        asm volatile("global_load_async_to_lds_b32 %0, %1, off"               \
                     :: "v"(lr_), "v"(gr_) : "memory");                       \
        asm volatile("global_load_async_to_lds_b32 %0, %1, off"               \
                     :: "v"(lc_), "v"(gc_) : "memory");                       \
        asm volatile("global_load_async_to_lds_b32 %0, %1, off"               \
                     :: "v"(lv_), "v"(gv_) : "memory");                       \
    } while (0)

    long base      = gw * 32;
    int  cur       = 0;
    bool cur_async = (base < (long)E) && ((long)E - base >= 32);
    if (cur_async) ISSUE_ASYNC(0, base);

    for (; base < (long)E; base += stride) {
        const long nextb      = base + stride;
        const bool next_async = (nextb < (long)E) && ((long)E - nextb >= 32);
        if (next_async) ISSUE_ASYNC(cur ^ 1, nextb);

        const int cnt = (int)(((long)E - base < 32) ? ((long)E - base) : 32);

        if (cur_async) {
            // Drain current chunk's 3 copies; allow next chunk's 3 in flight.
            if (next_async) {
                asm volatile("s_wait_asynccnt 3" ::: "memory");
            } else {
                asm volatile("s_wait_asynccnt 0" ::: "memory");
            }
            // Explicit LDS read-back of this lane's staged edge metadata.
            const unsigned ar = (unsigned)(uintptr_t)&s_meta[cur][wid][0][lane];
            const unsigned ac = (unsigned)(uintptr_t)&s_meta[cur][wid][1][lane];
            const unsigned av = (unsigned)(uintptr_t)&s_meta[cur][wid][2][lane];
            int r_l, c_l; unsigned v_u;
            asm volatile("ds_load_b32 %0, %3\n\t"
                         "ds_load_b32 %1, %4\n\t"
                         "ds_load_b32 %2, %5\n\t"
                         "s_wait_dscnt 0"
                         : "=&v"(r_l), "=&v"(c_l), "=&v"(v_u)
                         : "v"(ar), "v"(ac), "v"(av)
                         : "memory");

#pragma unroll 8
            for (int e = 0; e < 32; ++e) {
                const int   r = __builtin_amdgcn_readlane(r_l, e);
                const int   c = __builtin_amdgcn_readlane(c_l, e);
                const float v = __uint_as_float(
                    __builtin_amdgcn_readlane(v_u, e));
                const float4 x =
                    *(const float4*)(embs + (size_t)c * D_FEAT + lane * 4);
                float* o = out + (size_t)r * D_FEAT + lane * 4;
                __hip_atomic_fetch_add(o + 0, x.x * v, __ATOMIC_RELAXED,
                                       __HIP_MEMORY_SCOPE_AGENT);
                __hip_atomic_fetch_add(o + 1, x.y * v, __ATOMIC_RELAXED,
                                       __HIP_MEMORY_SCOPE_AGENT);
                __hip_atomic_fetch_add(o + 2, x.z * v, __ATOMIC_RELAXED,
                                       __HIP_MEMORY_SCOPE_AGENT);
                __hip_atomic_fetch_add(o + 3, x.w * v, __ATOMIC_RELAXED,
                                       __HIP_MEMORY_SCOPE_AGENT);
            }
        } else {
            // Partial tail chunk: wave-uniform metadata reads (SMEM path).
            for (int e = 0; e < cnt; ++e) {
                const int   r = erow[base + e];
                const int   c = ecol[base + e];
                const float v = eval[base + e];
                const float4 x =
                    *(const float4*)(embs + (size_t)c * D_FEAT + lane * 4);
                float* o = out + (size_t)r * D_FEAT + lane * 4;
                __hip_atomic_fetch_add(o + 0, x.x * v, __ATOMIC_RELAXED,
                                       __HIP_MEMORY_SCOPE_AGENT);
                __hip_atomic_fetch_add(o + 1, x.y * v, __ATOMIC_RELAXED,
                                       __HIP_MEMORY_SCOPE_AGENT);
                __hip_atomic_fetch_add(o + 2, x.z * v, __ATOMIC_RELAXED,
                                       __HIP_MEMORY_SCOPE_AGENT);
                __hip_atomic_fetch_add(o + 3, x.w * v, __ATOMIC_RELAXED,
                                       __HIP_MEMORY_SCOPE_AGENT);
            }
        }
        cur ^= 1;
        cur_async = next_async;
    }
#undef ISSUE_ASYNC
}

// ---------------------------------------------------------------------------
// Launch
// ---------------------------------------------------------------------------
extern "C" void kernel_launch(void* const* d_in, const int* in_sizes, int n_in,
                              void* d_out, int out_size, void* d_ws, size_t ws_size,
                              hipStream_t stream) {
    const int*   erow = (const int*)d_in[0];
    const int*   ecol = (const int*)d_in[1];
    const float* eval = (const float*)d_in[2];
    const float* embs = (const float*)d_in[3];
    float*       out  = (float*)d_out;

    const int E = in_sizes[0];

    long n4 = (long)out_size / 4;
    int  zb = (int)((n4 + 255) / 256);
    if (zb > 4096) zb = 4096;
    if (zb < 1) zb = 1;
    gcn_zero_out<<<zb, 256, 0, stream>>>((float4*)out, n4, out, (long)out_size);

    long edgeWaves = ((long)E + 31) / 32;
    int  blocks    = (int)((edgeWaves + 7) / 8);
    if (blocks > 4096) blocks = 4096;
    if (blocks < 1) blocks = 1;
    gcn_spmm_scatter<<<blocks, 256, 0, stream>>>(erow, ecol, eval, embs, out, E);
}